// Net_17446157156503
// MI455X (gfx1250) — compile-verified
//
#include <hip/hip_runtime.h>

typedef __attribute__((ext_vector_type(2))) float v2f;
typedef __attribute__((ext_vector_type(8))) float v8f;

#define BETA      0.95f
#define THRESH    1.0f
#define NUM_STEPS 25
#define INP  784
#define HID  1000
#define OUT  10
#define BATCH 4096
#define HID_PAD 1024          // padded stride for cur1 (64-divisible; pad cols never read)
#define MT 256                // fc1 M tiles (BATCH / 16)
#define NSUPER (HID_PAD / 64) // 16 super-tiles of 64 columns

#define WAVES_PER_BLOCK 2
#define ROWS_PER_BLOCK  (WAVES_PER_BLOCK * 16)

// D = A(16x4 f32) * B(4x16 f32) + C(16x16 f32), wave32
static __device__ __forceinline__ v8f wmma_f32x4(v2f a, v2f b, v8f c) {
    return __builtin_amdgcn_wmma_f32_16x16x4_f32(
        /*neg_a=*/false, a, /*neg_b=*/false, b,
        /*c_mod=*/(short)0, c, /*reuse_a=*/false, /*reuse_b=*/false);
}

// Fused LIF mem1 update on a (k, k+1) pair: read old mem1 from LDS, apply
// leak + input - subtract-reset, write back, return spike pair as WMMA A data.
static __device__ __forceinline__ v2f lif_pair(const float* __restrict__ c1p,
                                               float* __restrict__ m1p) {
    v2f c1 = *(const v2f*)c1p;
    v2f mo = *(v2f*)m1p;
    v2f mn;
    mn.x = BETA * mo.x + c1.x - ((mo.x > THRESH) ? THRESH : 0.0f);
    mn.y = BETA * mo.y + c1.y - ((mo.y > THRESH) ? THRESH : 0.0f);
    *(v2f*)m1p = mn;
    v2f a;
    a.x = (mn.x > THRESH) ? 1.0f : 0.0f;
    a.y = (mn.y > THRESH) ? 1.0f : 0.0f;
    return a;
}

// ---------------------------------------------------------------------------
// Kernel 1: cur1[b][h] = x[b][:] . W1[h][:] + b1[h]   (stride 1024)
// One wave computes a 16x64 tile: one A fragment feeds 4 independent WMMA
// accumulator chains. Pad columns (n >= 1000) use a CLAMPED W1 row pointer:
// loads stay in-bounds, results are garbage, but the step kernel never reads
// h >= 1000 -> no per-lane EXEC masking anywhere in the inner loop.
// ---------------------------------------------------------------------------
__global__ void fc1_kernel(const float* __restrict__ x,
                           const float* __restrict__ W1,
                           const float* __restrict__ b1,
                           float* __restrict__ cur1) {
    const int wgid = blockIdx.x * (blockDim.x >> 5) + (threadIdx.x >> 5);
    if (wgid >= MT * NSUPER) return;
    const int mt = wgid / NSUPER, nt = wgid % NSUPER;
    const int m0 = mt * 16, n0 = nt * 64;
    const int lane  = threadIdx.x & 31;
    const int lrow  = lane & 15;       // M for A, N for B/C
    const int khalf = lane >> 4;       // selects K pair {0,1} vs {2,3}

    const float* xrow = x + (size_t)(m0 + lrow) * INP + khalf * 2;
    const float* wr[4];
    #pragma unroll
    for (int s = 0; s < 4; ++s) {
        int n = n0 + s * 16 + lrow;
        if (n > HID - 1) n = HID - 1;                  // clamp: safe garbage
        wr[s] = W1 + (size_t)n * INP + khalf * 2;
    }

    v8f acc[4] = {};
    #pragma unroll 2
    for (int kc = 0; kc < INP / 4; ++kc) {
        const int off = kc * 4;
        v2f a   = *(const v2f*)(xrow  + off);
        v2f b0  = *(const v2f*)(wr[0] + off);
        v2f b1v = *(const v2f*)(wr[1] + off);
        v2f b2v = *(const v2f*)(wr[2] + off);
        v2f b3  = *(const v2f*)(wr[3] + off);
        acc[0] = wmma_f32x4(a, b0,  acc[0]);
        acc[1] = wmma_f32x4(a, b1v, acc[1]);
        acc[2] = wmma_f32x4(a, b2v, acc[2]);
        acc[3] = wmma_f32x4(a, b3,  acc[3]);
    }

    #pragma unroll
    for (int s = 0; s < 4; ++s) {
        const int n = n0 + s * 16 + lrow;
        const float bias = b1[(n < HID) ? n : (HID - 1)];
        #pragma unroll
        for (int v = 0; v < 8; ++v) {
            const int m = m0 + v + khalf * 8;          // C layout: M = v (+8 hi lanes)
            cur1[(size_t)m * HID_PAD + n] = acc[s][v] + bias;
        }
    }
}

// ---------------------------------------------------------------------------
// Kernel 2: fused 25-step LIF loop. Each wave owns 16 batch rows.
// mem1 (16 rows x 1000) lives in LDS for the whole kernel; mem1 update,
// spike threshold, and fc2 A-fragment generation are fused per K-chunk.
// The K=1000 fc2 reduction runs as TWO independent WMMA accumulator chains
// (even/odd 4-wide chunks) so the WMMA pipe always has an independent op in
// flight; chains are summed once per step. cur2 lives in the C registers;
// mem2 recurrence + outputs are register-resident in the natural C layout
// (M = v + 8*khalf, N = lane%16).
// ---------------------------------------------------------------------------
__global__ void snn_step_kernel(const float* __restrict__ cur1,
                                const float* __restrict__ W2,
                                const float* __restrict__ b2,
                                float* __restrict__ spk_out,
                                float* __restrict__ mem_out) {
    extern __shared__ float smem[];
    float* mem1 = smem;                                   // WAVES_PER_BLOCK*16*HID
    float* w2s  = smem + WAVES_PER_BLOCK * 16 * HID;      // 16*HID, rows >= OUT zeroed

    const int tid = threadIdx.x;
    for (int i = tid; i < WAVES_PER_BLOCK * 16 * HID; i += blockDim.x)
        mem1[i] = 0.0f;
    for (int i = tid; i < 16 * HID; i += blockDim.x) {
        const int nn = i / HID, kk = i % HID;
        w2s[i] = (nn < OUT) ? W2[nn * HID + kk] : 0.0f;
    }
    __syncthreads();

    const int wave  = tid >> 5;
    const int lane  = tid & 31;
    const int lrow  = lane & 15;
    const int khalf = lane >> 4;
    const int rbase = blockIdx.x * ROWS_PER_BLOCK + wave * 16;

    const float* c1row = cur1 + (size_t)(rbase + lrow) * HID_PAD + khalf * 2;
    float*       m1row = mem1 + wave * 16 * HID + lrow * HID + khalf * 2;
    const float* w2row = w2s + lrow * HID + khalf * 2;
    const float  bias2 = (lrow < OUT) ? b2[lrow] : 0.0f;

    v8f mem2 = {};
    for (int step = 0; step < NUM_STEPS; ++step) {
        v8f acc0 = {}, acc1 = {};
        #pragma unroll 2
        for (int kc = 0; kc < HID / 8; ++kc) {          // 125 iterations of 8 wide
            const int off = kc * 8;
            // even chunk -> chain 0
            v2f aE = lif_pair(c1row + off, m1row + off);
            v2f bE = *(const v2f*)(w2row + off);
            acc0 = wmma_f32x4(aE, bE, acc0);
            // odd chunk -> chain 1 (independent of chain 0)
            v2f aO = lif_pair(c1row + off + 4, m1row + off + 4);
            v2f bO = *(const v2f*)(w2row + off + 4);
            acc1 = wmma_f32x4(aO, bO, acc1);
        }
        // mem2 recurrence + outputs (C layout: M = v + 8*khalf, N = lrow)
        #pragma unroll
        for (int v = 0; v < 8; ++v) {
            const float cur2 = (acc0[v] + acc1[v]) + bias2;
            const float mo = mem2[v];
            const float mn = BETA * mo + cur2 - ((mo > THRESH) ? THRESH : 0.0f);
            mem2[v] = mn;
            if (lrow < OUT) {
                const int row = rbase + v + khalf * 8;
                const size_t idx = (size_t)step * (BATCH * OUT) + (size_t)row * OUT + lrow;
                spk_out[idx] = (mn > THRESH) ? 1.0f : 0.0f;
                mem_out[idx] = mn;
            }
        }
    }
}

extern "C" void kernel_launch(void* const* d_in, const int* in_sizes, int n_in,
                              void* d_out, int out_size, void* d_ws, size_t ws_size,
                              hipStream_t stream) {
    const float* x  = (const float*)d_in[0];
    const float* W1 = (const float*)d_in[1];
    const float* b1 = (const float*)d_in[2];
    const float* W2 = (const float*)d_in[3];
    const float* b2 = (const float*)d_in[4];

    float* cur1    = (float*)d_ws;                        // BATCH * HID_PAD floats
    float* spk_out = (float*)d_out;                       // [25,4096,10]
    float* mem_out = (float*)d_out + (size_t)NUM_STEPS * BATCH * OUT;

    // fc1: 256*16 = 4096 wave-tiles (16x64 each), 8 waves per 256-thread block
    dim3 blk1(256);
    dim3 grd1((MT * NSUPER + 7) / 8);
    fc1_kernel<<<grd1, blk1, 0, stream>>>(x, W1, b1, cur1);

    // fused step loop: 128 blocks * 2 waves, 192000 B dynamic LDS per block
    const size_t smem_bytes =
        (size_t)(WAVES_PER_BLOCK * 16 * HID + 16 * HID) * sizeof(float);
    dim3 blk2(WAVES_PER_BLOCK * 32);
    dim3 grd2(BATCH / ROWS_PER_BLOCK);
    snn_step_kernel<<<grd2, blk2, smem_bytes, stream>>>(cur1, W2, b2, spk_out, mem_out);
}